// Attention_82781199663345
// MI455X (gfx1250) — compile-verified
//
#include <hip/hip_runtime.h>
#include <math.h>

// Problem constants (match reference)
#define H_   1024
#define MD_  2048
#define L_   16384
#define NH_  8
#define DH_  128   // H/NH

typedef __attribute__((ext_vector_type(2))) float v2f;
typedef __attribute__((ext_vector_type(8))) float v8f;

static __device__ __forceinline__ v8f wmma_f32(v2f a, v2f b, v8f c) {
  // V_WMMA_F32_16X16X4_F32 : D(16x16,f32) = A(16x4,f32) x B(4x16,f32) + C
  return __builtin_amdgcn_wmma_f32_16x16x4_f32(
      /*neg_a=*/false, a, /*neg_b=*/false, b,
      /*c_mod=*/(short)0, c, /*reuse_a=*/false, /*reuse_b=*/false);
}

// ---------------------------------------------------------------------------
// K1: q = x @ Wq.T + bq, pre-scaled by d^-0.5.  qs[j*8+n] = Q[n,j].
__global__ void k_qproj(const float* __restrict__ x, const float* __restrict__ Wq,
                        const float* __restrict__ bq, float* __restrict__ qs) {
  int i = blockIdx.x * blockDim.x + threadIdx.x;           // 0..H-1
  const float* w = Wq + (size_t)i * H_;
  float acc = 0.f;
#pragma unroll 4
  for (int j = 0; j < H_; ++j) acc = fmaf(x[j], w[j], acc);
  const float scale = 0.08838834764831845f;                // 1/sqrt(128)
  qs[i] = (acc + bq[i]) * scale;
}

// ---------------------------------------------------------------------------
// K2: R[m][n] = sum_j Q[n,j] * Wk[j*8+n, m]  (n>=8 padded with 0), and zero aT.
// Rbuf stored K-pair interleaved: Rp[(m>>1)*32 + n*2 + (m&1)] so that the
// B-fragment (two K-adjacent values, same n) is ONE global_load_b64 in K3.
__global__ void k_rproj(const float* __restrict__ qs, const float* __restrict__ Wk,
                        float* __restrict__ Rbuf, float* __restrict__ aT) {
  int m = blockIdx.x * blockDim.x + threadIdx.x;           // 0..MD-1
  for (int n = 0; n < 16; ++n) {
    float acc = 0.f;
    if (n < NH_) {
#pragma unroll 4
      for (int j = 0; j < DH_; ++j)
        acc = fmaf(qs[j * NH_ + n], Wk[(size_t)(j * NH_ + n) * MD_ + m], acc);
    }
    Rbuf[(m >> 1) * 32 + n * 2 + (m & 1)] = acc;           // pair-interleaved
    aT[m * 16 + n] = 0.f;                                  // zero accumulator buffer
  }
}

// ---------------------------------------------------------------------------
// K3: scoresT[n][l] = memory[l,:] . R[:,n]   via WMMA f32 16x16x4.
// One wave per 16-row tile of memory; K-loop over MD.
__global__ void __launch_bounds__(32) k_scores(const float* __restrict__ mem,
                                               const float* __restrict__ Rbuf,
                                               float* __restrict__ scores) {
  const int lane = threadIdx.x;
  const int mrow = lane & 15;          // M index (memory row in tile) / N index
  const int h    = lane >> 4;          // lane half: fragment K = 2h + v
  const int l0   = blockIdx.x * 16;

  const float* arow = mem  + (size_t)(l0 + mrow) * MD_ + 2 * h;  // A: mem[l0+mrow][k+2h+v]
  const float* bpair = Rbuf + (size_t)h * 32 + 2 * mrow;         // pair (k/2+h, n=mrow)

  v8f acc = {};
  for (int k = 0; k < MD_; k += 4) {
    if ((k & 255) == 0) __builtin_prefetch(arow + 256, 0, 0);    // global_prefetch_b8
    v2f a = *(const v2f*)arow;                                   // one b64 load
    v2f b = *(const v2f*)bpair;                                  // one b64 load
    acc = wmma_f32(a, b, acc);
    arow  += 4;
    bpair += 64;                                                 // 2 K-pairs * 32
  }
  // D element (M = v + 8h, N = mrow) -> scoresT[n][l0 + v + 8h]; keep heads 0..7
  if (mrow < NH_) {
    float* srow = scores + (size_t)mrow * L_ + l0 + 8 * h;
#pragma unroll
    for (int v = 0; v < 8; ++v) srow[v] = acc[v];
  }
}

// ---------------------------------------------------------------------------
// K4: per-head softmax over L (adds bias c_n = Q[n,:].bk[head n]); in-place.
// (Reference pad-mask term is exactly 0.0f: TINY underflows in float32.)
__global__ void __launch_bounds__(256) k_softmax(const float* __restrict__ qs,
                                                 const float* __restrict__ bk,
                                                 float* __restrict__ scores) {
  const int n = blockIdx.x;            // head 0..7
  const int t = threadIdx.x;
  __shared__ float red[256];

  float c = 0.f;                       // bias fold (redundant per-thread, trivial)
  for (int j = 0; j < DH_; ++j) c = fmaf(qs[j * NH_ + n], bk[j * NH_ + n], c);

  float* row = scores + (size_t)n * L_;
  float m = -3.4e38f;
  for (int l = t; l < L_; l += 256) m = fmaxf(m, row[l] + c);
  red[t] = m; __syncthreads();
  for (int s = 128; s > 0; s >>= 1) { if (t < s) red[t] = fmaxf(red[t], red[t + s]); __syncthreads(); }
  m = red[0]; __syncthreads();

  float sum = 0.f;
  for (int l = t; l < L_; l += 256) sum += __expf(row[l] + c - m);
  red[t] = sum; __syncthreads();
  for (int s = 128; s > 0; s >>= 1) { if (t < s) red[t] += red[t + s]; __syncthreads(); }
  float inv = 1.0f / red[0];

  for (int l = t; l < L_; l += 256) row[l] = __expf(row[l] + c - m) * inv;
}

// ---------------------------------------------------------------------------
// K5: aT[m][n] = sum_l attn[n][l] * memory[l][m]  via WMMA (A = memory^T tile).
// Grid: (MD/16) x (L-chunks); chunks accumulate with global f32 atomics.
// Second streaming pass over `memory` — L2-resident (128 MB < 192 MB L2).
#define KC_ (L_ / 8)
__global__ void __launch_bounds__(32) k_wsum(const float* __restrict__ mem,
                                             const float* __restrict__ attn,
                                             float* __restrict__ aT) {
  const int lane = threadIdx.x;
  const int n16  = lane & 15;          // M (memory column) / N (head) index
  const int h    = lane >> 4;
  const int m0   = blockIdx.x * 16;
  const int kc   = blockIdx.y * KC_;

  const float* acol = mem + (size_t)(kc + 2 * h) * MD_ + m0 + n16;  // coalesced rows
  const bool   act  = n16 < NH_;
  const float  msk  = act ? 1.f : 0.f;
  const float* brow = attn + (size_t)(act ? n16 : 0) * L_ + kc + 2 * h;

  v8f acc = {};
  for (int k = 0; k < KC_; k += 4) {
    if ((k & 63) == 0) __builtin_prefetch(acol + (size_t)64 * MD_, 0, 0);
    v2f a; a.x = acol[0]; a.y = acol[MD_];      // A[v]: memory[kc+k+2h+v][m0+n16]
    v2f b = *(const v2f*)brow;                  // one b64 load (8B aligned)
    b.x *= msk; b.y *= msk;                     // zero pad-heads n>=8
    acc = wmma_f32(a, b, acc);
    acol += (size_t)4 * MD_;
    brow += 4;
  }
#pragma unroll
  for (int v = 0; v < 8; ++v)
    atomicAdd(&aT[(size_t)(m0 + v + 8 * h) * 16 + n16], acc[v]);    // global_atomic_add_f32
}

// ---------------------------------------------------------------------------
// K6: feat[j*8+n] = aT[:,n] . Wv[j*8+n,:] + bv[j*8+n]
__global__ void k_feat(const float* __restrict__ aT, const float* __restrict__ Wv,
                       const float* __restrict__ bv, float* __restrict__ feat) {
  int i = blockIdx.x * blockDim.x + threadIdx.x;   // 0..H-1, i = j*8+n
  int n = i & 7;
  const float* w = Wv + (size_t)i * MD_;
  float acc = 0.f;
#pragma unroll 4
  for (int m = 0; m < MD_; ++m) acc = fmaf(aT[m * 16 + n], w[m], acc);
  feat[i] = acc + bv[i];
}

// ---------------------------------------------------------------------------
// K7: out = relu(concat(x, feat) @ Wo.T + bo)
__global__ void k_out(const float* __restrict__ x, const float* __restrict__ feat,
                      const float* __restrict__ Wo, const float* __restrict__ bo,
                      float* __restrict__ out) {
  int i = blockIdx.x * blockDim.x + threadIdx.x;   // 0..H-1
  const float* w = Wo + (size_t)i * (2 * H_);
  float acc = 0.f;
#pragma unroll 4
  for (int j = 0; j < H_; ++j) acc = fmaf(x[j], w[j], acc);
#pragma unroll 4
  for (int j = 0; j < H_; ++j) acc = fmaf(feat[j], w[H_ + j], acc);
  acc += bo[i];
  out[i] = acc > 0.f ? acc : 0.f;
}

// ---------------------------------------------------------------------------
extern "C" void kernel_launch(void* const* d_in, const int* in_sizes, int n_in,
                              void* d_out, int out_size, void* d_ws, size_t ws_size,
                              hipStream_t stream) {
  const float* x   = (const float*)d_in[0];
  const float* mem = (const float*)d_in[1];
  const float* Wq  = (const float*)d_in[2];
  const float* bq  = (const float*)d_in[3];
  const float* Wk  = (const float*)d_in[4];
  const float* bk  = (const float*)d_in[5];
  const float* Wv  = (const float*)d_in[6];
  const float* bv  = (const float*)d_in[7];
  const float* Wo  = (const float*)d_in[8];
  const float* bo  = (const float*)d_in[9];
  float* out = (float*)d_out;

  // Workspace layout (floats): total ~0.78 MB
  float* ws     = (float*)d_ws;
  float* qs     = ws;                    // 1024
  float* Rbuf   = qs   + H_;             // MD*16 = 32768 (pair-interleaved)
  float* scores = Rbuf + MD_ * 16;       // 8*L  = 131072 (attn in-place)
  float* aT     = scores + NH_ * L_;     // MD*16 = 32768
  float* feat   = aT + MD_ * 16;         // 1024

  k_qproj  <<<H_ / 256, 256, 0, stream>>>(x, Wq, bq, qs);
  k_rproj  <<<MD_ / 256, 256, 0, stream>>>(qs, Wk, Rbuf, aT);
  k_scores <<<L_ / 16, 32, 0, stream>>>(mem, Rbuf, scores);
  k_softmax<<<NH_, 256, 0, stream>>>(qs, bk, scores);
  k_wsum   <<<dim3(MD_ / 16, L_ / KC_), 32, 0, stream>>>(mem, scores, aT);
  k_feat   <<<H_ / 256, 256, 0, stream>>>(aT, Wv, bv, feat);
  k_out    <<<H_ / 256, 256, 0, stream>>>(x, feat, Wo, bo, out);
}